// FastFoodWrap_78726750536120
// MI455X (gfx1250) — compile-verified
//
#include <hip/hip_runtime.h>
#include <hip/hip_bf16.h>

typedef float v2f __attribute__((ext_vector_type(2)));
typedef float v8f __attribute__((ext_vector_type(8)));

#define D_IN   4096
#define D_OUT  4096
#define BATCH  256
#define LLW    16777216   // 2^24
#define LLB    4096

// ---------------------------------------------------------------------------
// Kernel 0: single block. Computes
//   t_w = FWHT_1024(BB_w[:1024] * V)           -> ws
//   t_b = FWHT_1024(BB_b[:1024] * V)           (LDS only)
//   sum_b = sum(GG_b^2)                        (LDS reduce)
//   b = b0 + FWHT_4096(t_b[Pi_b & 1023]*GG_b) / sqrt(4096*sum_b)  -> ws
// ---------------------------------------------------------------------------
__global__ __launch_bounds__(1024) void prep_kernel(
    const float* __restrict__ V, const float* __restrict__ BB_w,
    const float* __restrict__ b0, const float* __restrict__ BB_b,
    const float* __restrict__ GG_b, const int* __restrict__ Pi_b,
    float* __restrict__ t_w_out, float* __restrict__ b_out)
{
    __shared__ float sht[1024];
    __shared__ float shm[4096];
    __shared__ float red[1024];
    const int tid = threadIdx.x;

    // ---- t_w ----
    sht[tid] = BB_w[tid] * V[tid];
    for (int s = 512; s >= 1; s >>= 1) {
        __syncthreads();
        if (tid < 512) {
            int r1 = ((tid & ~(s - 1)) << 1) | (tid & (s - 1));
            int r2 = r1 + s;
            float a = sht[r1], b = sht[r2];
            sht[r1] = a + b; sht[r2] = a - b;
        }
    }
    __syncthreads();
    t_w_out[tid] = sht[tid];
    __syncthreads();

    // ---- t_b ----
    sht[tid] = BB_b[tid] * V[tid];
    for (int s = 512; s >= 1; s >>= 1) {
        __syncthreads();
        if (tid < 512) {
            int r1 = ((tid & ~(s - 1)) << 1) | (tid & (s - 1));
            int r2 = r1 + s;
            float a = sht[r1], b = sht[r2];
            sht[r1] = a + b; sht[r2] = a - b;
        }
    }
    __syncthreads();

    // ---- sum(GG_b^2), deterministic tree ----
    float sb = 0.f;
    for (int k = 0; k < 4; ++k) {
        float g = GG_b[tid + k * 1024];
        sb += g * g;
    }
    red[tid] = sb;
    __syncthreads();
    for (int off = 512; off > 0; off >>= 1) {
        if (tid < off) red[tid] += red[tid + off];
        __syncthreads();
    }
    const float scale_b = rsqrtf(4096.0f * red[0]);

    // ---- gather + FWHT_4096 + scale + b0 ----
    for (int k = 0; k < 4; ++k) {
        int i = tid + k * 1024;
        shm[i] = sht[Pi_b[i] & 1023] * GG_b[i];
    }
    for (int s = 2048; s >= 1; s >>= 1) {
        __syncthreads();
        for (int j = tid; j < 2048; j += 1024) {
            int r1 = ((j & ~(s - 1)) << 1) | (j & (s - 1));
            int r2 = r1 + s;
            float a = shm[r1], b = shm[r2];
            shm[r1] = a + b; shm[r2] = a - b;
        }
    }
    __syncthreads();
    for (int k = 0; k < 4; ++k) {
        int i = tid + k * 1024;
        b_out[i] = b0[i] + scale_b * shm[i];
    }
}

// ---------------------------------------------------------------------------
// Kernel 1: m2[i] = t_w[Pi_w[i] & 1023] * GG_w[i]; per-block partial of GG^2.
// 16384 blocks x 256 threads x 4 elements.
// ---------------------------------------------------------------------------
__global__ __launch_bounds__(256) void gather_kernel(
    const int* __restrict__ Pi, const float* __restrict__ GG,
    const float* __restrict__ t, float* __restrict__ m2,
    float* __restrict__ partial)
{
    __shared__ float tsh[1024];
    __shared__ float red[256];
    const int tid = threadIdx.x;
    for (int j = tid; j < 1024; j += 256) tsh[j] = t[j];
    __syncthreads();

    const size_t base = (size_t)blockIdx.x * 1024;
    float s = 0.f;
    for (int k = 0; k < 4; ++k) {
        size_t i = base + (size_t)(k * 256 + tid);
        float g = GG[i];
        int   p = Pi[i] & 1023;
        m2[i] = tsh[p] * g;
        s += g * g;
    }
    red[tid] = s;
    __syncthreads();
    for (int off = 128; off > 0; off >>= 1) {
        if (tid < off) red[tid] += red[tid + off];
        __syncthreads();
    }
    if (tid == 0) partial[blockIdx.x] = red[0];
}

// ---------------------------------------------------------------------------
// Kernel 2: deterministic reduce of 16384 partials -> scale_w
// ---------------------------------------------------------------------------
__global__ __launch_bounds__(256) void reduce_scale_kernel(
    const float* __restrict__ partial, float* __restrict__ scale_out)
{
    __shared__ float red[256];
    const int tid = threadIdx.x;
    float s = 0.f;
    for (int k = 0; k < 64; ++k) s += partial[tid + k * 256];
    red[tid] = s;
    __syncthreads();
    for (int off = 128; off > 0; off >>= 1) {
        if (tid < off) red[tid] += red[tid + off];
        __syncthreads();
    }
    if (tid == 0) scale_out[0] = rsqrtf(16777216.0f * red[0]);
}

// ---------------------------------------------------------------------------
// Kernel 3: column FWHT (strides 4096..2^23 of the 2^24 transform).
// Block = 16 columns, 4096 rows staged in ~272KB dynamic LDS (CDNA5 320KB WGP LDS).
// ---------------------------------------------------------------------------
__global__ __launch_bounds__(1024) void col_fwht_kernel(float* __restrict__ m2)
{
    extern __shared__ float tile[];   // 4096 * 17 floats, pitch-17 to dodge bank conflicts
    const int tid = threadIdx.x;
    const int c0  = blockIdx.x * 16;

    for (int q = tid; q < 65536; q += 1024) {
        int r = q >> 4, c = q & 15;
        tile[r * 17 + c] = m2[(size_t)r * 4096 + c0 + c];
    }
    for (int s = 2048; s >= 1; s >>= 1) {
        __syncthreads();
        for (int q = tid; q < 32768; q += 1024) {
            int c = q & 15;
            int p = q >> 4;
            int r1 = ((p & ~(s - 1)) << 1) | (p & (s - 1));
            int r2 = r1 + s;
            float a = tile[r1 * 17 + c], b = tile[r2 * 17 + c];
            tile[r1 * 17 + c] = a + b;
            tile[r2 * 17 + c] = a - b;
        }
    }
    __syncthreads();
    for (int q = tid; q < 65536; q += 1024) {
        int r = q >> 4, c = q & 15;
        m2[(size_t)r * 4096 + c0 + c] = tile[r * 17 + c];
    }
}

// ---------------------------------------------------------------------------
// Kernel 4: row FWHT (strides 1..2048) + scale + add W0, in place -> W
// ---------------------------------------------------------------------------
__global__ __launch_bounds__(1024) void row_fwht_w_kernel(
    float* __restrict__ m2, const float* __restrict__ W0,
    const float* __restrict__ scale_ptr)
{
    __shared__ float sh[4096];
    const int tid = threadIdx.x;
    const size_t base = (size_t)blockIdx.x * 4096;

    for (int j = tid; j < 4096; j += 1024) sh[j] = m2[base + j];
    for (int s = 2048; s >= 1; s >>= 1) {
        __syncthreads();
        for (int j = tid; j < 2048; j += 1024) {
            int r1 = ((j & ~(s - 1)) << 1) | (j & (s - 1));
            int r2 = r1 + s;
            float a = sh[r1], b = sh[r2];
            sh[r1] = a + b; sh[r2] = a - b;
        }
    }
    __syncthreads();
    const float scale = scale_ptr[0];
    for (int j = tid; j < 4096; j += 1024)
        m2[base + j] = W0[base + j] + scale * sh[j];
}

// ---------------------------------------------------------------------------
// Kernel 5: out = x @ W^T + b using V_WMMA_F32_16X16X4_F32.
// Block: 64(M) x 64(N) tile, 8 waves as 4x2 grid, each wave = 16x32 (two C frags).
// K chunked by 64 through LDS.
// ---------------------------------------------------------------------------
__global__ __launch_bounds__(256) void gemm_kernel(
    const float* __restrict__ x, const float* __restrict__ W,
    const float* __restrict__ bias, float* __restrict__ out)
{
    __shared__ float As[64 * 68];
    __shared__ float Bs[64 * 68];

    const int tid   = threadIdx.x;
    const int lane  = tid & 31;
    const int wave  = tid >> 5;
    const int wr    = wave >> 1;       // 0..3 (M)
    const int wc    = wave & 1;        // 0..1 (N)
    const int mBase = blockIdx.y * 64;
    const int nBase = blockIdx.x * 64;

    const int fRow  = lane & 15;       // M (for A) / N (for B) within frag
    const int khalf = lane >> 4;       // 0 or 1: K offset 0/2 per f32 16x16x4 layout

    v8f c0 = {};
    v8f c1 = {};

    for (int kb = 0; kb < D_IN; kb += 64) {
        __syncthreads();
        for (int i = 0; i < 4; ++i) {
            int q  = tid + i * 256;       // 0..1023
            int r  = q >> 4;              // row 0..63
            int c4 = (q & 15) << 2;       // 0..60 step 4
            float4 av = *(const float4*)(x + (size_t)(mBase + r) * D_IN + kb + c4);
            *(float4*)(As + r * 68 + c4) = av;
            float4 bv = *(const float4*)(W + (size_t)(nBase + r) * D_IN + kb + c4);
            *(float4*)(Bs + r * 68 + c4) = bv;
        }
        __syncthreads();

        const float* aRow  = As + (wr * 16 + fRow) * 68;
        const float* bRow0 = Bs + (wc * 32 + fRow) * 68;
        const float* bRow1 = Bs + (wc * 32 + 16 + fRow) * 68;

        for (int kk = 0; kk < 64; kk += 4) {
            v2f a  = *(const v2f*)(aRow  + kk + 2 * khalf);
            v2f b0 = *(const v2f*)(bRow0 + kk + 2 * khalf);
            v2f b1 = *(const v2f*)(bRow1 + kk + 2 * khalf);
            // D = A(16x4) * B(4x16) + C ; f32 full precision
            c0 = __builtin_amdgcn_wmma_f32_16x16x4_f32(
                     false, a, false, b0, (short)0, c0, false, false);
            c1 = __builtin_amdgcn_wmma_f32_16x16x4_f32(
                     false, a, false, b1, (short)0, c1, false, false);
        }
    }

    // Epilogue: C/D layout -> lane = N, VGPR v = M row (+8 for lanes 16-31)
    const int nCol = lane & 15;
    const int mOff = (lane >> 4) * 8;
    const float bb0 = bias[nBase + wc * 32 + nCol];
    const float bb1 = bias[nBase + wc * 32 + 16 + nCol];
    for (int v = 0; v < 8; ++v) {
        int m = mBase + wr * 16 + mOff + v;
        out[(size_t)m * D_OUT + nBase + wc * 32 + nCol]      = c0[v] + bb0;
        out[(size_t)m * D_OUT + nBase + wc * 32 + 16 + nCol] = c1[v] + bb1;
    }
}

// ---------------------------------------------------------------------------
// Host launch
// ---------------------------------------------------------------------------
extern "C" void kernel_launch(void* const* d_in, const int* in_sizes, int n_in,
                              void* d_out, int out_size, void* d_ws, size_t ws_size,
                              hipStream_t stream)
{
    const float* x    = (const float*)d_in[0];
    const float* V    = (const float*)d_in[1];
    const float* W0   = (const float*)d_in[2];
    const float* b0   = (const float*)d_in[3];
    const float* BB_w = (const float*)d_in[4];
    const float* GG_w = (const float*)d_in[5];
    const float* BB_b = (const float*)d_in[6];
    const float* GG_b = (const float*)d_in[7];
    const int*   Pi_w = (const int*)d_in[8];
    const int*   Pi_b = (const int*)d_in[9];
    float* out = (float*)d_out;

    float* ws      = (float*)d_ws;
    float* m2      = ws;                          // 16,777,216 f  (m2 -> W in place)
    float* t_w     = ws + (size_t)LLW;            // 1024 f
    float* bvec    = t_w + 1024;                  // 4096 f
    float* partial = bvec + 4096;                 // 16384 f
    float* scalew  = partial + 16384;             // 1 f

    prep_kernel<<<1, 1024, 0, stream>>>(V, BB_w, b0, BB_b, GG_b, Pi_b, t_w, bvec);
    gather_kernel<<<16384, 256, 0, stream>>>(Pi_w, GG_w, t_w, m2, partial);
    reduce_scale_kernel<<<1, 256, 0, stream>>>(partial, scalew);
    col_fwht_kernel<<<256, 1024, 4096 * 17 * sizeof(float), stream>>>(m2);
    row_fwht_w_kernel<<<4096, 1024, 0, stream>>>(m2, W0, scalew);
    gemm_kernel<<<dim3(D_OUT / 64, BATCH / 64), 256, 0, stream>>>(x, m2, bvec, out);
}